// GNN_17592186044987
// MI455X (gfx1250) — compile-verified
//
#include <hip/hip_runtime.h>
#include <hip/hip_bf16.h>

typedef __attribute__((ext_vector_type(2))) float v2f;
typedef __attribute__((ext_vector_type(8))) float v8f;

// ---------------------------------------------------------------------------
// Workspace layout (floats): [agg1: 2N][deg: N][acc: N][s: N][r: N]  = 6N*4 B
// ---------------------------------------------------------------------------

__device__ __forceinline__ float fast_rcp(float x) {
    return __builtin_amdgcn_rcpf(x);   // v_rcp_f32
}

// Butterfly add across lanes via ds_swizzle (group-of-32 XOR mode):
// offset = {0, xor_mask[4:0], or_mask=0, and_mask=0x1F}
template <int XOR_MASK>
__device__ __forceinline__ float swz_xor_add(float v) {
    int p = __builtin_amdgcn_ds_swizzle(__float_as_int(v), (XOR_MASK << 10) | 0x1F);
    return v + __int_as_float(p);
}

__global__ void gnn_zero_f32(float* __restrict__ p, long long n) {
    long long i = (long long)blockIdx.x * blockDim.x + threadIdx.x;
    long long stride = (long long)gridDim.x * blockDim.x;
    for (; i < n; i += stride) p[i] = 0.0f;
}

// Pass 1 (edge-parallel): agg1[dst] += x[src] (2 f32), deg[dst] += 1
__global__ void gnn_scatter_l1(const float2* __restrict__ x,
                               const int* __restrict__ src,
                               const int* __restrict__ dst,
                               float* __restrict__ agg1,
                               float* __restrict__ deg,
                               int E) {
    int e = blockIdx.x * blockDim.x + threadIdx.x;
    if (e >= E) return;
    int s = __builtin_nontemporal_load(&src[e]);   // streamed: NT hint, keep L2 for accumulators
    int d = __builtin_nontemporal_load(&dst[e]);
    float2 xv = x[s];
    unsafeAtomicAdd(&agg1[2 * d + 0], xv.x);
    unsafeAtomicAdd(&agg1[2 * d + 1], xv.y);
    unsafeAtomicAdd(&deg[d], 1.0f);
}

// Pass 2 (node-parallel, 16 nodes per wave):
//   pre = [mean(agg1), x] (16x4) @ [W1l ; W1r] (4x16)  via V_WMMA_F32_16X16X4_F32
//   h   = relu(pre + b1)
//   s[i] = W2l . h_i ,  r[i] = W2r . h_i   (ds_swizzle reduction, h never stored)
__global__ void gnn_layer1_wmma(const float* __restrict__ x,
                                const float* __restrict__ agg1,
                                const float* __restrict__ deg,
                                const float* __restrict__ W1l,  // [16][2]
                                const float* __restrict__ W1r,  // [16][2]
                                const float* __restrict__ b1,   // [16]
                                const float* __restrict__ W2l,  // [16]
                                const float* __restrict__ W2r,  // [16]
                                float* __restrict__ s_out,
                                float* __restrict__ r_out,
                                int N) {
    const int gtid = blockIdx.x * blockDim.x + threadIdx.x;
    const int wave = gtid >> 5;
    const int lane = threadIdx.x & 31;
    const int base = wave * 16;           // wave-uniform
    if (base >= N) return;                // whole wave exits together

    const int  m    = lane & 15;
    const int  node = base + m;
    const bool lo   = lane < 16;

    if (base + 16 > N) {
        // Scalar tail path (wave-uniform branch; unreachable for N % 16 == 0)
        if (lo && node < N) {
            float rdg = fast_rcp(fmaxf(deg[node], 1.0f));
            float a0 = agg1[2 * node + 0] * rdg, a1 = agg1[2 * node + 1] * rdg;
            float x0 = x[2 * node + 0], x1 = x[2 * node + 1];
            float sv = 0.0f, rv = 0.0f;
            for (int n = 0; n < 16; ++n) {
                float h = fmaxf(a0 * W1l[2 * n] + a1 * W1l[2 * n + 1] +
                                x0 * W1r[2 * n] + x1 * W1r[2 * n + 1] + b1[n], 0.0f);
                sv += h * W2l[n];
                rv += h * W2r[n];
            }
            s_out[node] = sv;
            r_out[node] = rv;
        }
        return;
    }

    // A matrix (16x4 f32): lanes 0-15 -> K=0,1 (mean agg); lanes 16-31 -> K=2,3 (x)
    const float* aptr = lo ? (agg1 + 2 * node) : (x + 2 * node);
    float rdg = lo ? fast_rcp(fmaxf(deg[node], 1.0f)) : 1.0f;   // v_rcp_f32, no div sequence
    v2f a;
    a.x = aptr[0] * rdg;
    a.y = aptr[1] * rdg;

    // B matrix (4x16 f32): Wc[k][n], k=0,1 -> W1l[n][k]; k=2,3 -> W1r[n][k-2]
    // VGPR0: lo=K0 / hi=K2 ; VGPR1: lo=K1 / hi=K3 ; N = lane & 15
    const float* wptr = lo ? W1l : W1r;
    v2f b;
    b.x = wptr[2 * m + 0];
    b.y = wptr[2 * m + 1];

    v8f c = {};
    // D = A x B + 0 : emits v_wmma_f32_16x16x4_f32
    c = __builtin_amdgcn_wmma_f32_16x16x4_f32(
        /*neg_a=*/false, a, /*neg_b=*/false, b,
        /*c_mod=*/(short)0, c, /*reuse_a=*/false, /*reuse_b=*/false);

    // C/D layout: VGPR v holds h[M][N] with M = v (lanes 0-15) or v+8 (16-31), N = lane&15
    const float bias = b1[m];
    const float w2l  = W2l[m];
    const float w2r  = W2r[m];

#pragma unroll
    for (int v = 0; v < 8; ++v) {
        float h  = fmaxf(c[v] + bias, 0.0f);   // relu(pre + b1)
        float sv = h * w2l;
        float rv = h * w2r;
        // reduce over the 16 lanes of each half-wave (xor <= 8 stays in-half)
        sv = swz_xor_add<1>(sv);  rv = swz_xor_add<1>(rv);
        sv = swz_xor_add<2>(sv);  rv = swz_xor_add<2>(rv);
        sv = swz_xor_add<4>(sv);  rv = swz_xor_add<4>(rv);
        sv = swz_xor_add<8>(sv);  rv = swz_xor_add<8>(rv);
        if (m == v) {
            int out_node = base + v + (lo ? 0 : 8);
            s_out[out_node] = sv;
            r_out[out_node] = rv;
        }
    }
}

// Pass 3 (edge-parallel): acc[dst] += s[src]  (single scalar per edge)
__global__ void gnn_scatter_l2(const float* __restrict__ s,
                               const int* __restrict__ src,
                               const int* __restrict__ dst,
                               float* __restrict__ acc,
                               int E) {
    int e = blockIdx.x * blockDim.x + threadIdx.x;
    if (e >= E) return;
    int si = __builtin_nontemporal_load(&src[e]);
    int di = __builtin_nontemporal_load(&dst[e]);
    unsafeAtomicAdd(&acc[di], s[si]);
}

// Pass 4 (node-parallel): out = sigmoid(acc/deg + r + b2)
__global__ void gnn_finalize(const float* __restrict__ acc,
                             const float* __restrict__ deg,
                             const float* __restrict__ r,
                             const float* __restrict__ b2,
                             float* __restrict__ out,
                             int N) {
    int i = blockIdx.x * blockDim.x + threadIdx.x;
    if (i >= N) return;
    float z = acc[i] * fast_rcp(fmaxf(deg[i], 1.0f)) + r[i] + b2[0];
    out[i] = fast_rcp(1.0f + __expf(-z));
}

extern "C" void kernel_launch(void* const* d_in, const int* in_sizes, int n_in,
                              void* d_out, int out_size, void* d_ws, size_t ws_size,
                              hipStream_t stream) {
    (void)n_in; (void)out_size; (void)ws_size;

    const float* x   = (const float*)d_in[0];   // [N,2]
    const int*   ei  = (const int*)d_in[1];     // [2,E] (int32: JAX x64 disabled)
    const float* W1l = (const float*)d_in[2];   // [16,2]
    const float* b1  = (const float*)d_in[3];   // [16]
    const float* W1r = (const float*)d_in[4];   // [16,2]
    const float* W2l = (const float*)d_in[5];   // [1,16]
    const float* b2  = (const float*)d_in[6];   // [1]
    const float* W2r = (const float*)d_in[7];   // [1,16]

    const int N = in_sizes[0] / 2;
    const int E = in_sizes[1] / 2;
    const int* src = ei;
    const int* dst = ei + E;

    float* ws   = (float*)d_ws;
    float* agg1 = ws;                       // 2N
    float* deg  = ws + 2LL * N;             // N
    float* acc  = ws + 3LL * N;             // N
    float* sbuf = ws + 4LL * N;             // N
    float* rbuf = ws + 5LL * N;             // N

    const int BT = 256;

    // Zero the accumulated buffers (agg1, deg, acc) every call (deterministic).
    gnn_zero_f32<<<2048, BT, 0, stream>>>(ws, 4LL * N);

    gnn_scatter_l1<<<(E + BT - 1) / BT, BT, 0, stream>>>(
        (const float2*)x, src, dst, agg1, deg, E);

    const int waves   = (N + 15) / 16;
    const int threads = waves * 32;
    gnn_layer1_wmma<<<(threads + BT - 1) / BT, BT, 0, stream>>>(
        x, agg1, deg, W1l, W1r, b1, W2l, W2r, sbuf, rbuf, N);

    gnn_scatter_l2<<<(E + BT - 1) / BT, BT, 0, stream>>>(sbuf, src, dst, acc, E);

    gnn_finalize<<<(N + BT - 1) / BT, BT, 0, stream>>>(
        acc, deg, rbuf, b2, (float*)d_out, N);
}